// CenterDisc_9079560864491
// MI455X (gfx1250) — compile-verified
//
#include <hip/hip_runtime.h>
#include <math.h>

typedef __attribute__((ext_vector_type(2))) float v2f;
typedef __attribute__((ext_vector_type(8))) float v8f;

#define NSAMP  4096
#define NFEAT  32768          // 64 * 512 features per sample
#define NCLS   4
#define CHUNK  1024           // samples per chunk
#define NCHUNK 4              // NSAMP / CHUNK
#define WAVES  8              // waves per block (256 threads, wave32)
#define NPAIR  6              // 4 choose 2

// ---------------------------------------------------------------------------
// Kernel 1: segment-sum of x over labels, expressed as WMMA F32 16x16x4.
//   D(16x16) = A(16x4, one-hot classes) x B(4x16, x tile) + C
// Each wave owns 16 features x one 1024-sample chunk and writes its partial
// class sums (4 x 16 floats) with plain stores -> fully deterministic.
// ---------------------------------------------------------------------------
__global__ __launch_bounds__(256) void seg_wmma(
    const float* __restrict__ x,            // (NSAMP, NFEAT) row-major
    const long long* __restrict__ labels,   // (NSAMP,) int64 in [0, NCLS)
    float* __restrict__ part)               // (NCHUNK, NCLS, NFEAT)
{
    __shared__ int slab[CHUNK];

    const int tid    = threadIdx.x;
    const int wave   = tid >> 5;
    const int lane   = tid & 31;
    const int lane16 = lane & 15;      // M row (class) for A, N col for B/D
    const int hi     = lane >> 4;      // 0: K=0,1  1: K=2,3 (A and B layout)
    const int chunk  = blockIdx.y;
    const int base   = chunk * CHUNK;

    // Stage this chunk's labels into LDS once per block (int64 -> int).
    for (int i = tid; i < CHUNK; i += 256)
        slab[i] = (int)labels[base + i];
    __syncthreads();

    const int f0 = (blockIdx.x * WAVES + wave) * 16;
    // Column pointer: element (row = base + 2*hi, col = f0 + lane16).
    const float* p = x + (size_t)(base + 2 * hi) * NFEAT + f0 + lane16;

    v8f acc = {0.f, 0.f, 0.f, 0.f, 0.f, 0.f, 0.f, 0.f};

#pragma unroll 4
    for (int idx = 0; idx < CHUNK; idx += 4) {
        // Stream-ahead prefetch (global_prefetch_b8); speculative, safe.
        __builtin_prefetch(p + (size_t)16 * NFEAT, 0, 0);

        // B 4x16: VGPR0 = rows K0/K2, VGPR1 = rows K1/K3 (split by lane half).
        v2f b;
        b.x = p[0];
        b.y = p[NFEAT];

        // A 16x4 one-hot: lane16 is the class row M, K follows same split.
        const int l0 = slab[idx + 2 * hi];
        const int l1 = slab[idx + 2 * hi + 1];
        v2f a;
        a.x = (l0 == lane16) ? 1.0f : 0.0f;
        a.y = (l1 == lane16) ? 1.0f : 0.0f;

        // v_wmma_f32_16x16x4_f32: D = A x B + C (fp32 accumulate, exact-ish).
        acc = __builtin_amdgcn_wmma_f32_16x16x4_f32(
            /*neg_a=*/false, a, /*neg_b=*/false, b,
            /*c_mod=*/(short)0, acc, /*reuse_a=*/false, /*reuse_b=*/false);

        p += (size_t)4 * NFEAT;
    }

    // D layout: VGPR c holds row M=c for lanes 0-15; classes 0..3 only.
    if (hi == 0) {
        float* dst = part + (size_t)chunk * NCLS * NFEAT + f0 + lane16;
        dst[(size_t)0 * NFEAT] = acc[0];
        dst[(size_t)1 * NFEAT] = acc[1];
        dst[(size_t)2 * NFEAT] = acc[2];
        dst[(size_t)3 * NFEAT] = acc[3];
    }
}

// ---------------------------------------------------------------------------
// Kernel 2 (single block): label histogram (exact int atomics), deterministic
// chunk reduction -> centers, 6 pairwise squared-Frobenius sums with a fixed-
// order LDS tree reduction, sqrt, mean over pairs -> scalar output.
// ---------------------------------------------------------------------------
__global__ __launch_bounds__(256) void finalize(
    const long long* __restrict__ labels,
    const float* __restrict__ part,         // (NCHUNK, NCLS, NFEAT)
    float* __restrict__ out)
{
    __shared__ int   cnt[NCLS];
    __shared__ float red[256];

    const int tid = threadIdx.x;

    if (tid < NCLS) cnt[tid] = 0;
    __syncthreads();
    for (int i = tid; i < NSAMP; i += 256)
        atomicAdd(&cnt[(int)labels[i]], 1);
    __syncthreads();

    float inv[NCLS];
#pragma unroll
    for (int c = 0; c < NCLS; ++c)
        inv[c] = 1.0f / fmaxf((float)cnt[c], 1.0f);

    float ss[NPAIR] = {0.f, 0.f, 0.f, 0.f, 0.f, 0.f};

    for (int f = tid; f < NFEAT; f += 256) {
        float ctr[NCLS];
#pragma unroll
        for (int c = 0; c < NCLS; ++c) {
            float s = 0.f;
#pragma unroll
            for (int ch = 0; ch < NCHUNK; ++ch)
                s += part[((size_t)ch * NCLS + c) * NFEAT + f];
            ctr[c] = s * inv[c];
        }
        int pi = 0;
#pragma unroll
        for (int i2 = 0; i2 < NCLS; ++i2)
#pragma unroll
            for (int j = i2 + 1; j < NCLS; ++j) {
                const float d = ctr[i2] - ctr[j];
                ss[pi++] += d * d;
            }
    }

    float tot = 0.f;
#pragma unroll
    for (int pi = 0; pi < NPAIR; ++pi) {
        red[tid] = ss[pi];
        __syncthreads();
        for (int s = 128; s > 0; s >>= 1) {
            if (tid < s) red[tid] += red[tid + s];
            __syncthreads();
        }
        if (tid == 0) tot += sqrtf(red[0]);
        __syncthreads();
    }

    if (tid == 0) out[0] = tot / (float)NPAIR;
}

// ---------------------------------------------------------------------------
extern "C" void kernel_launch(void* const* d_in, const int* in_sizes, int n_in,
                              void* d_out, int out_size, void* d_ws, size_t ws_size,
                              hipStream_t stream) {
    (void)in_sizes; (void)n_in; (void)out_size; (void)ws_size;

    const float*     x      = (const float*)d_in[0];
    const long long* labels = (const long long*)d_in[1];
    float*           part   = (float*)d_ws;   // NCHUNK*NCLS*NFEAT floats = 2 MB
    float*           out    = (float*)d_out;

    // 2048 feature tiles (16 cols each) / 8 waves per block = 256 blocks in x,
    // 4 sample chunks in y. Every (tile, chunk) partial is written each call,
    // so no scratch zeroing is needed and results are deterministic.
    dim3 grid(NFEAT / 16 / WAVES, NCHUNK);
    seg_wmma<<<grid, 256, 0, stream>>>(x, labels, part);
    finalize<<<1, 256, 0, stream>>>(labels, part, out);
}